// GaussianLayer_64003602645438
// MI455X (gfx1250) — compile-verified
//
#include <hip/hip_runtime.h>

// ---------------------------------------------------------------------------
// Gaussian region log-likelihood as per-region f32 GEMM on CDNA5 WMMA.
//   ll[b,r,k] = A[b,:] . W[r,:,k] + C[r,k]
//   A[b,:]  = [ x_g(b,0..15) , x_g(b,0..15)^2 ]           (B x 32)
//   W[r,d,k]     =  mu/sigma^2        (d  = 0..15)
//   W[r,16+d,k]  = -0.5/sigma^2
//   C[r,k]  = -0.5*sum((mu/sigma)^2) - sum(log sigma) - 0.5*D*log(2pi)
// Engine: V_WMMA_F32_16X16X4_F32 chained 8x over the 32-deep inner dim,
// two N-halves for K=32 components.  Memory-bound (~50MB @ 23.3TB/s ~2.2us).
// v2: k-major LDS layout (B frags via single ds_load_b64), B frags + bias
//     hoisted to registers, 4 batch tiles per block to amortize region prep,
//     nontemporal output stores (write-once stream).
// ---------------------------------------------------------------------------

typedef __attribute__((ext_vector_type(2))) float v2f;
typedef __attribute__((ext_vector_type(8))) float v8f;

#define F_LOG_2PI 1.8378770664093453f

constexpr int BATCH = 4096;
constexpr int NFEAT = 1024;
constexpr int NREG  = 64;
constexpr int NCOMP = 32;
constexpr int NDIM  = 16;
constexpr int WAVES = 8;              // waves per block (wave32)
constexpr int BM    = WAVES * 16;     // 128 batch rows per tile
constexpr int TILES = 4;              // batch tiles per block (amortize prep)

__global__ __launch_bounds__(256)
void gauss_ll_wmma(const float* __restrict__ x,
                   const long long* __restrict__ regions,
                   const float* __restrict__ means,
                   const float* __restrict__ scales,
                   float* __restrict__ out)
{
    // k-major: Wlds[n*32 + dd], n = component (0..31), dd = inner dim (0..31)
    __shared__ float Wlds[32 * 32];
    __shared__ float Clds[32];
    __shared__ int   idx[NDIM];

    const int r     = blockIdx.x;
    const int tile0 = blockIdx.y * (BM * TILES);
    const int tid   = threadIdx.x;
    const int lane  = tid & 31;
    const int wid   = tid >> 5;

    // --- Region prep (wave 0): build W[r] (32x32 k-major) and C[r,k] --------
    if (tid < NDIM) idx[tid] = (int)regions[(size_t)r * NDIM + tid];
    if (tid < NCOMP) {
        const int k = tid;
        const float* mp = means  + ((size_t)r * NCOMP + k) * NDIM;
        const float* sp = scales + ((size_t)r * NCOMP + k) * NDIM;
        float c = 0.0f;
        #pragma unroll
        for (int d = 0; d < NDIM; ++d) {
            const float mu  = mp[d];
            const float sg  = sp[d];
            const float inv = 1.0f / sg;
            const float i2  = inv * inv;
            Wlds[k * 32 + d]      = mu * i2;       // linear term
            Wlds[k * 32 + 16 + d] = -0.5f * i2;    // quadratic term
            const float ms = mu * inv;
            c += -0.5f * ms * ms - logf(sg);
        }
        Clds[k] = c - 0.5f * (float)NDIM * F_LOG_2PI;
    }
    __syncthreads();

    // --- Per-lane fragment coordinates (32-bit WMMA layouts) ----------------
    // A 16x4 f32: lanes 0-15 hold M=lane (K=0,1 in v0,v1); lanes 16-31 hold
    // M=lane-16 (K=2,3). B 4x16 f32 mirrors it: K = 2*(lane>>4)+v, N = lane&15.
    const int half = lane >> 4;
    const int m16  = lane & 15;

    // --- Hoist tile-invariant B fragments + bias into registers -------------
    // Lane's two K-values are contiguous in k-major layout -> one ds_load_b64.
    v2f bLo[8], bHi[8];
    #pragma unroll
    for (int kk = 0; kk < 8; ++kk) {
        const int ddb = kk * 4 + 2 * half;          // even -> 8B aligned
        bLo[kk] = *(const v2f*)&Wlds[m16 * 32 + ddb];
        bHi[kk] = *(const v2f*)&Wlds[(16 + m16) * 32 + ddb];
    }
    const float cLo = Clds[m16];
    const float cHi = Clds[16 + m16];
    int id8[8];
    #pragma unroll
    for (int q = 0; q < 4; ++q) {
        id8[q * 2 + 0] = idx[q * 4 + 2 * half + 0];
        id8[q * 2 + 1] = idx[q * 4 + 2 * half + 1];
    }

    // --- Batch tiles: gather A, 16 chained WMMAs, biased NT store -----------
    const int mbase = half ? 8 : 0;
    for (int t = 0; t < TILES; ++t) {
        const int tile = tile0 + t * BM;
        const int row  = tile + wid * 16 + m16;

        const float* xr = x + (size_t)row * NFEAT;
        float xv[4][2];
        #pragma unroll
        for (int q = 0; q < 4; ++q) {
            xv[q][0] = xr[id8[q * 2 + 0]];
            xv[q][1] = xr[id8[q * 2 + 1]];
        }

        v8f accLo = {};   // components k = 0..15
        v8f accHi = {};   // components k = 16..31
        #pragma unroll
        for (int kk = 0; kk < 8; ++kk) {
            v2f a;
            if (kk < 4) {                       // linear half: x
                a.x = xv[kk][0];
                a.y = xv[kk][1];
            } else {                            // quadratic half: x^2
                const float t0 = xv[kk - 4][0];
                const float t1 = xv[kk - 4][1];
                a.x = t0 * t0;
                a.y = t1 * t1;
            }
            accLo = __builtin_amdgcn_wmma_f32_16x16x4_f32(
                false, a, false, bLo[kk], (short)0, accLo, false, false);
            accHi = __builtin_amdgcn_wmma_f32_16x16x4_f32(
                false, a, false, bHi[kk], (short)0, accHi, false, false);
        }

        // D layout: lanes 0-15 N=lane, VGPR i -> M=i; lanes 16-31 -> M=8+i.
        float* op = out + (size_t)(tile + wid * 16) * (NREG * NCOMP)
                        + (size_t)r * NCOMP;
        #pragma unroll
        for (int i = 0; i < 8; ++i) {
            const size_t ro = (size_t)(mbase + i) * (NREG * NCOMP);
            __builtin_nontemporal_store(accLo[i] + cLo, &op[ro + m16]);
            __builtin_nontemporal_store(accHi[i] + cHi, &op[ro + 16 + m16]);
        }
    }
}

extern "C" void kernel_launch(void* const* d_in, const int* in_sizes, int n_in,
                              void* d_out, int out_size, void* d_ws, size_t ws_size,
                              hipStream_t stream) {
    (void)in_sizes; (void)n_in; (void)out_size; (void)d_ws; (void)ws_size;
    const float*     x       = (const float*)d_in[0];
    const long long* regions = (const long long*)d_in[1];   // int64 per reference
    const float*     means   = (const float*)d_in[2];
    const float*     scales  = (const float*)d_in[3];
    float*           out     = (float*)d_out;

    dim3 grid(NREG, BATCH / (BM * TILES));   // 64 regions x 8 tile-groups
    dim3 block(32 * WAVES);                  // 8 wave32 waves
    gauss_ll_wmma<<<grid, block, 0, stream>>>(x, regions, means, scales, out);
}